// GCN_79972291051707
// MI455X (gfx1250) — compile-verified
//
#include <hip/hip_runtime.h>

typedef __attribute__((ext_vector_type(2))) float v2f;
typedef __attribute__((ext_vector_type(8))) float v8f;

// ---------------------------------------------------------------- utilities
__global__ void zero_f32v4(float4* __restrict__ p, long long n4) {
    long long i = (long long)blockIdx.x * blockDim.x + threadIdx.x;
    if (i < n4) p[i] = make_float4(0.f, 0.f, 0.f, 0.f);
}

__global__ void degree_kernel(const int* __restrict__ src, const int* __restrict__ dst,
                              long long E, float* __restrict__ odeg, float* __restrict__ ideg) {
    long long e = (long long)blockIdx.x * blockDim.x + threadIdx.x;
    if (e < E) {
        atomicAdd(&odeg[src[e]], 1.0f);
        atomicAdd(&ideg[dst[e]], 1.0f);
    }
}

__global__ void norm_kernel(float* __restrict__ onorm, float* __restrict__ inorm, int N) {
    int i = blockIdx.x * blockDim.x + threadIdx.x;
    if (i < N) {
        onorm[i] = 1.0f / sqrtf(fmaxf(onorm[i], 1.0f));
        inorm[i] = 1.0f / sqrtf(fmaxf(inorm[i], 1.0f));
    }
}

// ------------------------------------------------- WMMA GEMM: T = (s.*A) @ W
// A: [N,128] fp32 row-major, W: [128,Kout] fp32 row-major, Kout <= NT*16.
// W is staged TRANSPOSED in LDS (Wt[n][k], padded stride 132 -> the b64 B
// fragment read is bank-conflict-free and lands in an aligned VGPR pair).
// 256 threads = 8 waves; wave w owns rows [blk*128 + w*16, +16); each wave
// accumulates NT 16x16 output tiles over K with V_WMMA_F32_16X16X4_F32.
template <int NT>
__launch_bounds__(256)
__global__ void gemm_rowscale_wmma(const float* __restrict__ A, const float* __restrict__ scale,
                                   const float* __restrict__ W, float* __restrict__ T,
                                   int N, int Kout) {
    constexpr int KIN     = 128;
    constexpr int WCOLS   = NT * 16;
    constexpr int KSTRIDE = KIN + 4;  // 132: (n*132+k) % 64 = (4*(n%16)+k) % 64
    __shared__ float Wt[WCOLS * KSTRIDE];

    // cooperative transpose-load of W (zero-pad cols [Kout, WCOLS))
    for (int idx = threadIdx.x; idx < KIN * WCOLS; idx += blockDim.x) {
        int k = idx / WCOLS, n = idx % WCOLS;
        Wt[n * KSTRIDE + k] = (n < Kout) ? W[k * Kout + n] : 0.0f;
    }
    __syncthreads();

    const int wave = threadIdx.x >> 5;
    const int lane = threadIdx.x & 31;
    const int half = lane >> 4;      // 0: K={k,k+1}, 1: K={k+2,k+3}
    const int mrow = lane & 15;
    const int row0 = (blockIdx.x * 8 + wave) * 16;
    const int row  = row0 + mrow;

    const float s = (row < N) ? scale[row] : 0.0f;
    const float* __restrict__ arow = A + (size_t)(row < N ? row : 0) * KIN;

    v8f acc[NT];
    {
        v8f z = {};
#pragma unroll
        for (int t = 0; t < NT; ++t) acc[t] = z;
    }

    for (int k0 = 0; k0 < KIN; k0 += 4) {
        const int ka = k0 + half * 2;
        // A fragment (16x4): lane holds row 'row', K = ka, ka+1 (pre-scaled)
        v2f a = *(const v2f*)(arow + ka);
        a.x *= s;
        a.y *= s;
#pragma unroll
        for (int t = 0; t < NT; ++t) {
            // B fragment (4x16): lane holds col t*16+mrow, K = ka, ka+1
            // contiguous in transposed LDS -> single ds_load_b64
            const v2f b = *(const v2f*)&Wt[(t * 16 + mrow) * KSTRIDE + ka];
            acc[t] = __builtin_amdgcn_wmma_f32_16x16x4_f32(
                false, a, false, b, (short)0, acc[t], false, false);
        }
    }

    // C/D layout: VGPR i -> M = i + half*8, N = mrow
#pragma unroll
    for (int t = 0; t < NT; ++t) {
        const int col = t * 16 + mrow;
        if (col < Kout) {
#pragma unroll
            for (int i = 0; i < 8; ++i) {
                const int r = row0 + half * 8 + i;
                if (r < N) T[(size_t)r * Kout + col] = acc[t][i];
            }
        }
    }
}

// -------------------------------------------------- edge scatter-accumulate
// one thread per (edge, 4-float chunk): agg[dst[e]] += T[src[e]]
__global__ void scatter_add(const float* __restrict__ T, const int* __restrict__ src,
                            const int* __restrict__ dst, float* __restrict__ agg,
                            long long E, int K) {
    const long long tid    = (long long)blockIdx.x * blockDim.x + threadIdx.x;
    const int       chunks = K >> 2;
    const long long total  = E * (long long)chunks;
    if (tid >= total) return;
    const long long e = tid / chunks;
    const int       c = (int)(tid - e * chunks) << 2;
    const int sN = src[e];
    const int dN = dst[e];
    const float4 v = *(const float4*)(T + (size_t)sN * K + c);
    float* o = agg + (size_t)dN * K + c;
    atomicAdd(o + 0, v.x);
    atomicAdd(o + 1, v.y);
    atomicAdd(o + 2, v.z);
    atomicAdd(o + 3, v.w);
}

// ---------------------------------------- out = [relu](agg*inorm + b), x4
__global__ void postproc_v4(const float4* __restrict__ agg, const float* __restrict__ inorm,
                            const float4* __restrict__ bias, float4* __restrict__ out,
                            long long total4, int K4, int do_relu) {
    long long idx = (long long)blockIdx.x * blockDim.x + threadIdx.x;
    if (idx >= total4) return;
    long long i = idx / K4;
    int       c = (int)(idx - i * K4);
    const float  s = inorm[i];
    const float4 a = agg[idx];
    const float4 b = bias[c];
    float4 v = make_float4(a.x * s + b.x, a.y * s + b.y, a.z * s + b.z, a.w * s + b.w);
    if (do_relu) {
        v.x = fmaxf(v.x, 0.f); v.y = fmaxf(v.y, 0.f);
        v.z = fmaxf(v.z, 0.f); v.w = fmaxf(v.w, 0.f);
    }
    out[idx] = v;
}

// ---------------------------------------------------------------- launcher
extern "C" void kernel_launch(void* const* d_in, const int* in_sizes, int n_in,
                              void* d_out, int out_size, void* d_ws, size_t ws_size,
                              hipStream_t stream) {
    const float* x   = (const float*)d_in[0];
    const int*   src = (const int*)d_in[1];
    const int*   dst = (const int*)d_in[2];
    const float* W1  = (const float*)d_in[3];
    const float* b1  = (const float*)d_in[4];
    const float* W2  = (const float*)d_in[5];
    const float* b2  = (const float*)d_in[6];
    const float* W3  = (const float*)d_in[7];
    const float* b3  = (const float*)d_in[8];

    const int       IN_F  = 128;
    const int       H_F   = 128;
    const int       N_CLS = in_sizes[8];                 // 40
    const long long N     = (long long)in_sizes[0] / IN_F;
    const long long E     = in_sizes[1];

    float* ws    = (float*)d_ws;
    float* onorm = ws;  ws += N;
    float* inorm = ws;  ws += N;
    float* t     = ws;  ws += N * H_F;
    float* agg   = ws;  ws += N * H_F;
    float* h     = ws;  ws += N * H_F;

    auto cdiv = [](long long a, long long b) -> int { return (int)((a + b - 1) / b); };

    // degrees -> symmetric norms (deg counts exact in fp32; order independent)
    zero_f32v4<<<cdiv(2 * N / 4, 256), 256, 0, stream>>>((float4*)onorm, 2 * N / 4);
    degree_kernel<<<cdiv(E, 256), 256, 0, stream>>>(src, dst, E, onorm, inorm);
    norm_kernel<<<cdiv(N, 256), 256, 0, stream>>>(onorm, inorm, (int)N);

    const int gGemm = cdiv(N, 128);

    // ---- layer 1: h = relu(inorm .* segsum_dst((onorm.*x)@W1 [src]) + b1)
    gemm_rowscale_wmma<8><<<gGemm, 256, 0, stream>>>(x, onorm, W1, t, (int)N, H_F);
    zero_f32v4<<<cdiv(N * H_F / 4, 256), 256, 0, stream>>>((float4*)agg, N * H_F / 4);
    scatter_add<<<cdiv(E * (H_F / 4), 256), 256, 0, stream>>>(t, src, dst, agg, E, H_F);
    postproc_v4<<<cdiv(N * H_F / 4, 256), 256, 0, stream>>>((const float4*)agg, inorm,
        (const float4*)b1, (float4*)h, N * H_F / 4, H_F / 4, 1);

    // ---- layer 2
    gemm_rowscale_wmma<8><<<gGemm, 256, 0, stream>>>(h, onorm, W2, t, (int)N, H_F);
    zero_f32v4<<<cdiv(N * H_F / 4, 256), 256, 0, stream>>>((float4*)agg, N * H_F / 4);
    scatter_add<<<cdiv(E * (H_F / 4), 256), 256, 0, stream>>>(t, src, dst, agg, E, H_F);
    postproc_v4<<<cdiv(N * H_F / 4, 256), 256, 0, stream>>>((const float4*)agg, inorm,
        (const float4*)b2, (float4*)h, N * H_F / 4, H_F / 4, 1);

    // ---- layer 3 (40 classes, no relu) -> d_out
    gemm_rowscale_wmma<3><<<gGemm, 256, 0, stream>>>(h, onorm, W3, t, (int)N, N_CLS);
    zero_f32v4<<<cdiv(N * N_CLS / 4, 256), 256, 0, stream>>>((float4*)agg, N * N_CLS / 4);
    scatter_add<<<cdiv(E * (N_CLS / 4), 256), 256, 0, stream>>>(t, src, dst, agg, E, N_CLS);
    postproc_v4<<<cdiv(N * N_CLS / 4, 256), 256, 0, stream>>>((const float4*)agg, inorm,
        (const float4*)b3, (float4*)d_out, N * N_CLS / 4, N_CLS / 4, 0);
}